// DifferentiableCBFLayer_84791244357684
// MI455X (gfx1250) — compile-verified
//
#include <hip/hip_runtime.h>
#include <stdint.h>

// ---- problem constants (match reference) ----
#define MAX_OBS  10
#define MAX_NEI  7
#define N_POWER  30
#define N_FISTA  300
#define BLOCK    256

#define A_MAX    2.0f
#define W_MAX    1.0f
#define D_OBS2   0.25f     // 0.5^2
#define D_SAFE2  0.64f     // 0.8^2
#define D_MAX2   25.0f     // 5^2
#define DAMP     3.0f      // G_C1+G_C2 (same for A and CN)
#define STIFF    2.0f      // G_C1*G_C2
#define BIGC     1000.0f
#define PSL      0.005f    // 1/(2*W_SLACK)
#define ND       25        // dual dim: 3 slack + 10 obs + 7 nei + 1 cn + 4 box (dead row dropped: exactly zero contribution)

typedef unsigned int v4u __attribute__((ext_vector_type(4)));
typedef int          v8i __attribute__((ext_vector_type(8)));
typedef int          v4i __attribute__((ext_vector_type(4)));

#if defined(__HIP_DEVICE_COMPILE__) && __has_builtin(__builtin_amdgcn_tensor_load_to_lds)
#define USE_TDM 1
#else
#define USE_TDM 0
#endif

__global__ __launch_bounds__(BLOCK, 1)
void cbf_fista_kernel(const float* __restrict__ u_nom,   // (B,2)
                      const float* __restrict__ v_cur,   // (B,1)
                      const float* __restrict__ p_obs,   // (B,10,2)
                      const float* __restrict__ obs_m,   // (B,10)
                      const float* __restrict__ p_ag,    // (B,7,2)
                      const float* __restrict__ v_ag,    // (B,7,2)
                      const float* __restrict__ ag_m,    // (B,7)
                      const float* __restrict__ p_c,     // (B,1,2)
                      const float* __restrict__ v_c,     // (B,1,2)
                      const float* __restrict__ c_m,     // (B,1)
                      float* __restrict__ out,           // (B,2)
                      int n)
{
    // Single LDS object -> TDM lds_addr offset 0.
    __shared__ __align__(16) float s_pobs[BLOCK * 2 * MAX_OBS];   // 20 KB

    const int tid = threadIdx.x;
    const int b   = blockIdx.x * BLOCK + tid;
    const unsigned remain = (unsigned)((n - blockIdx.x * BLOCK) < BLOCK ? (n - blockIdx.x * BLOCK) : BLOCK);
    const unsigned elems  = remain * 2u * MAX_OBS;                // <= 5120, fits tile_dim0 (16b)

#if USE_TDM
    // ---- Tensor Data Mover: stage this block's p_obs slab into LDS ----
    if (tid == 0) {  // only wave 0 issues (TDM ignores EXEC; other waves skip the branch)
        const uint64_t ga = (uint64_t)(uintptr_t)(p_obs + (uint64_t)blockIdx.x * (BLOCK * 2u * MAX_OBS));
        v4u g0; v8i g1; v4i g2 = {0, 0, 0, 0}; v4i g3 = {0, 0, 0, 0};
        v8i g4 = {0, 0, 0, 0, 0, 0, 0, 0};
        // D# group0: count=1 | lds_addr=0 | global_addr | type=2
        g0.x = 1u;
        g0.y = 0u;
        g0.z = (unsigned)(ga & 0xFFFFFFFFu);
        g0.w = (unsigned)((ga >> 32) & 0x1FFFFFFu) | 0x80000000u;
        // D# group1: data_size=4B, 1 row of `elems` elements, stride=elems
        g1[0] = (int)(2u << 16);                                  // data_size=2 (4B), mask/pad/iterate=0
        g1[1] = (int)((elems & 0xFFFFu) << 16);                   // tensor_dim0[15:0]
        g1[2] = (int)((elems >> 16) | (1u << 16));                // tensor_dim0[31:16] | tensor_dim1=1
        g1[3] = (int)((elems & 0xFFFFu) << 16);                   // tile_dim0 = elems
        g1[4] = (int)1;                                           // tile_dim1=1, tile_dim2=0
        g1[5] = (int)elems;                                       // tensor_dim0_stride[31:0]
        g1[6] = 0; g1[7] = 0;
        __builtin_amdgcn_tensor_load_to_lds(g0, g1, g2, g3, g4, 0);
#if __has_builtin(__builtin_amdgcn_s_wait_tensorcnt)
        __builtin_amdgcn_s_wait_tensorcnt(0);
#else
        asm volatile("s_wait_tensorcnt 0x0" ::: "memory");
#endif
    }
    __syncthreads();
#else
    if (b < n) {
        #pragma unroll
        for (int k = 0; k < 2 * MAX_OBS; ++k)
            s_pobs[tid * 2 * MAX_OBS + k] = p_obs[(size_t)b * 2 * MAX_OBS + k];
    }
    __syncthreads();
#endif

    if (b >= n) return;

    // ---------------- build sparse row data (all in VGPRs) ----------------
    const float u0 = u_nom[2 * b], u1 = u_nom[2 * b + 1];
    const float v  = v_cur[b];

    float a0[MAX_OBS], a1[MAX_OBS], mo[MAX_OBS], bo[MAX_OBS];
    #pragma unroll
    for (int i = 0; i < MAX_OBS; ++i) {
        const float lx = s_pobs[tid * 2 * MAX_OBS + 2 * i];
        const float ly = s_pobs[tid * 2 * MAX_OBS + 2 * i + 1];
        const float m  = obs_m[b * MAX_OBS + i];
        const float h   = lx * lx + ly * ly - D_OBS2;
        const float hd  = -2.0f * lx * v;
        const float rhs = 2.0f * v * v + DAMP * hd + STIFF * h;
        a0[i] = 2.0f * lx * m;
        a1[i] = 2.0f * ly * v * m;
        mo[i] = m;
        bo[i] = (m > 0.0f) ? rhs : BIGC;
    }

    float n0[MAX_NEI], n1[MAX_NEI], mn[MAX_NEI], bn[MAX_NEI];
    #pragma unroll
    for (int j = 0; j < MAX_NEI; ++j) {
        const float ax  = p_ag[b * 2 * MAX_NEI + 2 * j];
        const float ay  = p_ag[b * 2 * MAX_NEI + 2 * j + 1];
        const float vjx = v_ag[b * 2 * MAX_NEI + 2 * j];
        const float vjy = v_ag[b * 2 * MAX_NEI + 2 * j + 1];
        const float m   = ag_m[b * MAX_NEI + j];
        const float h   = ax * ax + ay * ay - D_SAFE2;
        const float hd  = -2.0f * ax * v + 2.0f * (ax * vjx + ay * vjy);
        const float hdd = 2.0f * v * v - 2.0f * v * vjx + 2.0f * (-v * vjx + vjx * vjx + vjy * vjy);
        const float rhs = hdd + DAMP * hd + STIFF * h;
        n0[j] = 2.0f * ax * m;
        n1[j] = (2.0f * ay * v - 2.0f * ay * vjx + 2.0f * ax * vjy) * m;
        mn[j] = m;
        bn[j] = (m > 0.0f) ? rhs : BIGC;
    }

    float c0, c1, mc, bc;
    {
        const float cx = p_c[2 * b], cy = p_c[2 * b + 1];
        const float cvx = v_c[2 * b], cvy = v_c[2 * b + 1];
        const float m = c_m[b];
        const float h   = D_MAX2 - (cx * cx + cy * cy);
        const float hd  = -(-2.0f * cx * v + 2.0f * (cx * cvx + cy * cvy));
        const float hdd = -(2.0f * v * v - 2.0f * v * cvx + 2.0f * (-v * cvx + cvx * cvx + cvy * cvy));
        const float rhs = hdd + DAMP * hd + STIFF * h;
        c0 = -2.0f * cx * m;
        c1 = -(2.0f * cy * v - 2.0f * cy * cvx + 2.0f * cx * cvy) * m;
        mc = m;
        bc = (m > 0.0f) ? rhs : BIGC;
    }

    // Aᵀz for the sparse structure (z indexed: 0..2 slack, 3..12 obs, 13..19 nei, 20 cn, 21..24 box)
    #define AT_Z(z, t0, t1, t2, t3, t4)                                        \
        {                                                                      \
            t0 = 0.0f; t1 = 0.0f; t2 = -(z)[0]; t3 = -(z)[1]; t4 = -(z)[2];    \
            _Pragma("unroll")                                                  \
            for (int i = 0; i < MAX_OBS; ++i) {                                \
                t0 += a0[i] * (z)[3 + i];                                      \
                t1 += a1[i] * (z)[3 + i];                                      \
                t2 -= mo[i] * (z)[3 + i];                                      \
            }                                                                  \
            _Pragma("unroll")                                                  \
            for (int j = 0; j < MAX_NEI; ++j) {                                \
                t0 += n0[j] * (z)[13 + j];                                     \
                t1 += n1[j] * (z)[13 + j];                                     \
                t3 -= mn[j] * (z)[13 + j];                                     \
            }                                                                  \
            t0 += c0 * (z)[20];  t1 += c1 * (z)[20];  t4 -= mc * (z)[20];      \
            t0 += -(z)[21] + (z)[22];                                          \
            t1 += -(z)[23] + (z)[24];                                          \
        }

    // ---------------- power iteration for step = 1/(||A Pinv Aᵀ||_2 + 1e-6) ----------------
    float vv[ND];
    #pragma unroll
    for (int i = 0; i < ND; ++i) vv[i] = 1.0f;

    #pragma unroll 1
    for (int it = 0; it < N_POWER; ++it) {
        float t0, t1, t2, t3, t4;
        AT_Z(vv, t0, t1, t2, t3, t4);
        const float p0 = 0.5f * t0, p1 = 0.5f * t1;
        const float p2 = PSL * t2, p3 = PSL * t3, p4 = PSL * t4;
        float w[ND];
        w[0] = -p2; w[1] = -p3; w[2] = -p4;
        #pragma unroll
        for (int i = 0; i < MAX_OBS; ++i) w[3 + i] = a0[i] * p0 + a1[i] * p1 - mo[i] * p2;
        #pragma unroll
        for (int j = 0; j < MAX_NEI; ++j) w[13 + j] = n0[j] * p0 + n1[j] * p1 - mn[j] * p3;
        w[20] = c0 * p0 + c1 * p1 - mc * p4;
        w[21] = -p0; w[22] = p0; w[23] = -p1; w[24] = p1;
        float nrm2 = 0.0f;
        #pragma unroll
        for (int i = 0; i < ND; ++i) nrm2 += w[i] * w[i];
        const float inv = 1.0f / (sqrtf(nrm2) + 1e-12f);
        #pragma unroll
        for (int i = 0; i < ND; ++i) vv[i] = w[i] * inv;
    }
    float step;
    {
        float t0, t1, t2, t3, t4;
        AT_Z(vv, t0, t1, t2, t3, t4);
        const float L = 0.5f * (t0 * t0 + t1 * t1) + PSL * (t2 * t2 + t3 * t3 + t4 * t4);
        step = 1.0f / (L + 1e-6f);
    }

    // ---------------- FISTA on the dual ----------------
    float lam[ND], y[ND];
    #pragma unroll
    for (int i = 0; i < ND; ++i) { lam[i] = 0.0f; y[i] = 0.0f; }
    float tk = 1.0f;

    #pragma unroll 1
    for (int it = 0; it < N_FISTA; ++it) {
        float t0, t1, t2, t3, t4;
        AT_Z(y, t0, t1, t2, t3, t4);
        const float x0 = u0 - 0.5f * t0;     // -( -2u0 + t0 ) * 0.5
        const float x1 = u1 - 0.5f * t1;
        const float x2 = -PSL * t2, x3 = -PSL * t3, x4 = -PSL * t4;

        const float tk1  = 0.5f * (1.0f + sqrtf(1.0f + 4.0f * tk * tk));
        const float beta = (tk - 1.0f) / tk1;
        tk = tk1;

        float r[ND];
        r[0] = -x2; r[1] = -x3; r[2] = -x4;
        #pragma unroll
        for (int i = 0; i < MAX_OBS; ++i) r[3 + i]  = a0[i] * x0 + a1[i] * x1 - mo[i] * x2 - bo[i];
        #pragma unroll
        for (int j = 0; j < MAX_NEI; ++j) r[13 + j] = n0[j] * x0 + n1[j] * x1 - mn[j] * x3 - bn[j];
        r[20] = c0 * x0 + c1 * x1 - mc * x4 - bc;
        r[21] = -x0 - A_MAX; r[22] = x0 - A_MAX;
        r[23] = -x1 - W_MAX; r[24] = x1 - W_MAX;

        #pragma unroll
        for (int i = 0; i < ND; ++i) {
            const float ln = fmaxf(y[i] + step * r[i], 0.0f);
            y[i]   = ln + beta * (ln - lam[i]);
            lam[i] = ln;
        }
    }

    // final primal from lam
    {
        float t0, t1, t2, t3, t4;
        AT_Z(lam, t0, t1, t2, t3, t4);
        out[2 * b]     = u0 - 0.5f * t0;
        out[2 * b + 1] = u1 - 0.5f * t1;
    }
    #undef AT_Z
}

extern "C" void kernel_launch(void* const* d_in, const int* in_sizes, int n_in,
                              void* d_out, int out_size, void* d_ws, size_t ws_size,
                              hipStream_t stream) {
    (void)n_in; (void)d_ws; (void)ws_size; (void)out_size;
    const float* u_nom = (const float*)d_in[0];
    const float* v_cur = (const float*)d_in[1];
    const float* p_obs = (const float*)d_in[2];
    const float* obs_m = (const float*)d_in[3];
    const float* p_ag  = (const float*)d_in[4];
    const float* v_ag  = (const float*)d_in[5];
    const float* ag_m  = (const float*)d_in[6];
    const float* p_c   = (const float*)d_in[7];
    const float* v_c   = (const float*)d_in[8];
    const float* c_m   = (const float*)d_in[9];
    float* out = (float*)d_out;

    const int n = in_sizes[0] / 2;          // batch size
    const int blocks = (n + BLOCK - 1) / BLOCK;
    hipLaunchKernelGGL(cbf_fista_kernel, dim3(blocks), dim3(BLOCK), 0, stream,
                       u_nom, v_cur, p_obs, obs_m, p_ag, v_ag, ag_m, p_c, v_c, c_m, out, n);
}